// quan_bn_40750649704878
// MI455X (gfx1250) — compile-verified
//
#include <hip/hip_runtime.h>
#include <stdint.h>

// Problem constants (x: [64, 64, 112, 112] fp32, NCHW)
#define NB       64
#define CCH      64
#define HW       12544                 // 112*112
#define NPLANES  (NB*CCH)              // 4096 planes of HW floats
#define PLANE_BYTES (HW*4)             // 50176, 16B-aligned
#define THREADS  448                   // 14 waves; 448*4 floats = 1792 floats/chunk
#define CHUNK_BYTES (THREADS*16)       // 7168
#define NCHUNKS  7                     // 7*7168 = 50176 = one plane exactly
#define NWAVES   (THREADS/32)
#define DEPTH    4                     // async pipeline depth (circular LDS buffers)
#define CNT_F    802816.0f             // NB*HW elements per channel

typedef float f32x4 __attribute__((ext_vector_type(4)));

// ---- CDNA5 async-copy primitives (ASYNCcnt-tracked Global->LDS DMA) ----
__device__ __forceinline__ uint32_t lds_addr32(const void* p) {
    // AS(3) flat pointers: low 32 bits are the wave-relative LDS byte address
    return (uint32_t)(uintptr_t)p;
}
// Fused s_wait_dscnt makes the LDS WAR hazard (async engine overwriting a
// buffer with an in-flight ds_load) architecturally impossible, instead of
// relying on the compiler's placement of the data-dependence wait.
__device__ __forceinline__ void async_b128(uint32_t lds_off, uint32_t goff,
                                           unsigned long long sbase) {
    asm volatile("s_wait_dscnt 0x0\n\t"
                 "global_load_async_to_lds_b128 %0, %1, %2"
                 :: "v"(lds_off), "v"(goff), "s"(sbase) : "memory");
}
__device__ __forceinline__ void wait_async(int n) {   // n is compile-time const
    switch (n) {
    case 0:  asm volatile("s_wait_asynccnt 0x0" ::: "memory"); break;
    case 1:  asm volatile("s_wait_asynccnt 0x1" ::: "memory"); break;
    case 2:  asm volatile("s_wait_asynccnt 0x2" ::: "memory"); break;
    default: asm volatile("s_wait_asynccnt 0x3" ::: "memory"); break;
    }
}

// ---- Kernel 1: per-plane partial stats (sum, sumsq, min, max) ----
__global__ void __launch_bounds__(THREADS)
stats_kernel(const float* __restrict__ x, float* __restrict__ partials) {
    __shared__ float buf[DEPTH][THREADS * 4];   // 4 x 7168 B circular buffer
    __shared__ float red[4][NWAVES];

    const int tid = threadIdx.x;
    const int p   = blockIdx.x;                 // plane = n*CCH + c
    const unsigned long long base =
        (unsigned long long)(uintptr_t)x + (unsigned long long)p * PLANE_BYTES;

    uint32_t lds[DEPTH];
    #pragma unroll
    for (int j = 0; j < DEPTH; ++j) lds[j] = lds_addr32(&buf[j][tid * 4]);
    const uint32_t goff0 = (uint32_t)(tid * 16);

    #pragma unroll
    for (int j = 0; j < DEPTH - 1; ++j)         // prime chunks 0..2
        async_b128(lds[j], goff0 + (uint32_t)j * CHUNK_BYTES, base);

    float s = 0.0f, ss = 0.0f;
    float mn = __builtin_inff(), mx = -__builtin_inff();

    #pragma unroll
    for (int k = 0; k < NCHUNKS; ++k) {
        const int nxt = k + DEPTH - 1;
        if (nxt < NCHUNKS)
            async_b128(lds[nxt % DEPTH], goff0 + (uint32_t)nxt * CHUNK_BYTES, base);
        const int last = (nxt < NCHUNKS) ? nxt : (NCHUNKS - 1);
        wait_async(last - k);                   // chunk k has landed (in-order)

        const f32x4 v = *reinterpret_cast<const f32x4*>(&buf[k % DEPTH][tid * 4]);
        s += v.x; ss = fmaf(v.x, v.x, ss); mn = fminf(mn, v.x); mx = fmaxf(mx, v.x);
        s += v.y; ss = fmaf(v.y, v.y, ss); mn = fminf(mn, v.y); mx = fmaxf(mx, v.y);
        s += v.z; ss = fmaf(v.z, v.z, ss); mn = fminf(mn, v.z); mx = fmaxf(mx, v.z);
        s += v.w; ss = fmaf(v.w, v.w, ss); mn = fminf(mn, v.w); mx = fmaxf(mx, v.w);
    }

    // wave32 shuffle-xor tree
    #pragma unroll
    for (int off = 16; off >= 1; off >>= 1) {
        s  += __shfl_xor(s,  off, 32);
        ss += __shfl_xor(ss, off, 32);
        mn  = fminf(mn, __shfl_xor(mn, off, 32));
        mx  = fmaxf(mx, __shfl_xor(mx, off, 32));
    }
    const int wave = tid >> 5;
    if ((tid & 31) == 0) {
        red[0][wave] = s; red[1][wave] = ss; red[2][wave] = mn; red[3][wave] = mx;
    }
    __syncthreads();
    if (tid == 0) {
        float S = red[0][0], SS = red[1][0], MN = red[2][0], MX = red[3][0];
        #pragma unroll
        for (int w = 1; w < NWAVES; ++w) {
            S += red[0][w]; SS += red[1][w];
            MN = fminf(MN, red[2][w]); MX = fmaxf(MX, red[3][w]);
        }
        const int c = p & (CCH - 1);
        const int n = p >> 6;
        const int idx = c * NB + n;             // [stat][c][n] layout
        partials[0 * NPLANES + idx] = S;
        partials[1 * NPLANES + idx] = SS;
        partials[2 * NPLANES + idx] = MN;
        partials[3 * NPLANES + idx] = MX;
    }
}

// ---- Kernel 2: finalize per-channel constants {a, b, qp0, qp1, qp2} ----
__global__ void __launch_bounds__(64)
finalize_kernel(const float* __restrict__ partials,
                const float* __restrict__ gamma,
                const float* __restrict__ beta,
                float* __restrict__ cvals) {
    __shared__ float red[4][2];
    const int c = blockIdx.x;
    const int t = threadIdx.x;                  // 0..63 over n

    float s  = partials[0 * NPLANES + c * NB + t];
    float ss = partials[1 * NPLANES + c * NB + t];
    float mn = partials[2 * NPLANES + c * NB + t];
    float mx = partials[3 * NPLANES + c * NB + t];

    #pragma unroll
    for (int off = 16; off >= 1; off >>= 1) {
        s  += __shfl_xor(s,  off, 32);
        ss += __shfl_xor(ss, off, 32);
        mn  = fminf(mn, __shfl_xor(mn, off, 32));
        mx  = fmaxf(mx, __shfl_xor(mx, off, 32));
    }
    if ((t & 31) == 0) {
        const int w = t >> 5;
        red[0][w] = s; red[1][w] = ss; red[2][w] = mn; red[3][w] = mx;
    }
    __syncthreads();
    if (t == 0) {
        const float S  = red[0][0] + red[0][1];
        const float SS = red[1][0] + red[1][1];
        const float MN = fminf(red[2][0], red[2][1]);
        const float MX = fmaxf(red[3][0], red[3][1]);

        const float mean    = S / CNT_F;
        const float var     = SS / CNT_F - mean * mean;
        const float inv_std = rsqrtf(var + 1e-5f);
        const float g  = gamma[c];
        const float be = beta[c];
        const float a  = g * inv_std;           // fake = a*x + b
        const float b  = be - mean * a;
        const float range = MX - MN;
        const float bm = 0.5f * range;
        const float bv = sqrtf(range);
        const float com = bm - bv * be / g;

        float* o = cvals + c * 8;
        o[0] = a;
        o[1] = b;
        o[2] = bv * (0.5f / 3.0f) / g + com;    // thresholds ascending
        o[3] = bv * (1.5f / 3.0f) / g + com;
        o[4] = bv * (2.5f / 3.0f) / g + com;
    }
}

// ---- Kernel 3: elementwise quantize + STE output ----
__global__ void __launch_bounds__(THREADS)
quant_kernel(const float* __restrict__ x, const float* __restrict__ cvals,
             float* __restrict__ out) {
    __shared__ float buf[DEPTH][THREADS * 4];

    const int tid = threadIdx.x;
    // Reverse plane order vs pass 1: the tail of x is still L2-resident.
    const int p = (NPLANES - 1) - (int)blockIdx.x;
    const int c = p & (CCH - 1);

    const float a  = cvals[c * 8 + 0];
    const float b  = cvals[c * 8 + 1];
    const float q0 = cvals[c * 8 + 2];
    const float q1 = cvals[c * 8 + 3];
    const float q2 = cvals[c * 8 + 4];

    const unsigned long long base =
        (unsigned long long)(uintptr_t)x + (unsigned long long)p * PLANE_BYTES;
    f32x4* __restrict__ outv = reinterpret_cast<f32x4*>(out + (size_t)p * HW);

    uint32_t lds[DEPTH];
    #pragma unroll
    for (int j = 0; j < DEPTH; ++j) lds[j] = lds_addr32(&buf[j][tid * 4]);
    const uint32_t goff0 = (uint32_t)(tid * 16);

    #pragma unroll
    for (int j = 0; j < DEPTH - 1; ++j)
        async_b128(lds[j], goff0 + (uint32_t)j * CHUNK_BYTES, base);

    #pragma unroll
    for (int k = 0; k < NCHUNKS; ++k) {
        const int nxt = k + DEPTH - 1;
        if (nxt < NCHUNKS)
            async_b128(lds[nxt % DEPTH], goff0 + (uint32_t)nxt * CHUNK_BYTES, base);
        const int last = (nxt < NCHUNKS) ? nxt : (NCHUNKS - 1);
        wait_async(last - k);

        const f32x4 v = *reinterpret_cast<const f32x4*>(&buf[k % DEPTH][tid * 4]);

        f32x4 r;
        #pragma unroll
        for (int e = 0; e < 4; ++e) {
            const float xv = v[e];
            const float fake = fmaf(a, xv, b);
            const float cnt = (xv > q0 ? 1.0f : 0.0f)
                            + (xv > q1 ? 1.0f : 0.0f)
                            + (xv > q2 ? 1.0f : 0.0f);
            // cnt * RN(1/3) == cnt/3 exactly for cnt in {0,1,2}; fix cnt==3.
            float q = cnt * (1.0f / 3.0f);
            if (cnt == 3.0f) q = 1.0f;
            r[e] = (q - fake) + fake;           // STE: value == q + fp noise
        }
        // out is write-once: NT store keeps it from evicting x's tail in L2.
        __builtin_nontemporal_store(r, outv + (k * THREADS + tid));
    }
}

extern "C" void kernel_launch(void* const* d_in, const int* in_sizes, int n_in,
                              void* d_out, int out_size, void* d_ws, size_t ws_size,
                              hipStream_t stream) {
    (void)in_sizes; (void)n_in; (void)out_size; (void)ws_size;
    const float* x     = (const float*)d_in[0];
    const float* gamma = (const float*)d_in[1];
    const float* beta  = (const float*)d_in[2];
    float* out = (float*)d_out;

    float* ws       = (float*)d_ws;
    float* partials = ws;                       // 4 * 4096 floats
    float* cvals    = ws + 4 * NPLANES;         // 64 * 8 floats

    stats_kernel   <<<NPLANES, THREADS, 0, stream>>>(x, partials);
    finalize_kernel<<<CCH,     64,      0, stream>>>(partials, gamma, beta, cvals);
    quant_kernel   <<<NPLANES, THREADS, 0, stream>>>(x, cvals, out);
}